// PCEN_25434796327076
// MI455X (gfx1250) — compile-verified
//
#include <hip/hip_runtime.h>
#include <hip/hip_bf16.h>

// PCEN (per-channel energy normalization) for x[64,256,2048,1] fp32.
// IIR smoother along T done as an SSD-style chunked scan using
// V_WMMA_F32_16X16X4_F32: per 16-row x 16-step tile,
//   M_tile = X_tile(16x16) * L(16x16) + a^(t+1) * carry
// with L[k][n] = s * a^(n-k) for k<=n (constant, held per-lane).
// Memory-bound kernel: ~256 MiB traffic -> ~11us floor @ 23.3 TB/s.

typedef float v2f __attribute__((ext_vector_type(2)));
typedef float v8f __attribute__((ext_vector_type(8)));

#define T_LEN   2048
#define SMOOTH  0.015f
#define DECAY   (1.0f - SMOOTH)
#define EPS_F   1e-9f

// hardware transcendentals: v_log_f32 (log2), v_exp_f32 (2^x)
#define LOG2F(v)  __builtin_amdgcn_logf(v)
#define EXP2F(v)  __builtin_amdgcn_exp2f(v)

// exact-ish integer power by squaring (branchless, init-time only)
__device__ __forceinline__ float powi_dec(int e) {
    float r = 1.0f, p = DECAY;
    #pragma unroll
    for (int b = 0; b < 5; ++b) {
        if (e & (1 << b)) r *= p;
        p *= p;
    }
    return r;
}

__global__ __launch_bounds__(256) void pcen_wmma_kernel(
    const float* __restrict__ x,
    const float* __restrict__ alpha_p,
    const float* __restrict__ power_p,
    const float* __restrict__ bias_p,
    float* __restrict__ out,
    int nrows)
{
    const int lane  = threadIdx.x & 31;
    const int wave  = threadIdx.x >> 5;
    const int group = blockIdx.x * 8 + wave;   // one wave owns 16 rows
    const int rbase = group * 16;
    if (rbase >= nrows) return;

    const int half = lane >> 4;   // 0: lanes 0-15, 1: lanes 16-31
    const int l16  = lane & 15;

    const float alpha    = alpha_p[0];
    const float power    = power_p[0];
    const float bias     = bias_p[0];
    const float bias_pow = EXP2F(power * LOG2F(bias));

    // ---- Constant B operand: L chunks in WMMA B layout -------------------
    // B chunk kc, element j of the per-lane float2 holds
    //   K = kc*4 + j + 2*half  (row of L), N = l16 (column of L)
    //   L[K][N] = s * a^(N-K) if K<=N else 0
    v2f bmat[4];
    #pragma unroll
    for (int kc = 0; kc < 4; ++kc) {
        #pragma unroll
        for (int j = 0; j < 2; ++j) {
            int kg = kc * 4 + j + 2 * half;
            int e  = l16 - kg;
            bmat[kc][j] = (e >= 0) ? (SMOOTH * powi_dec(e)) : 0.0f;
        }
    }
    // carry weight: a^(n+1) with n = time-in-tile = l16
    const float apc = powi_dec(l16 + 1);

    // ---- Per-lane address offsets ---------------------------------------
    const float* xrow = x   + (size_t)rbase * T_LEN;
    float*       orow = out + (size_t)rbase * T_LEN;

    // A-operand (16x4 f32 layout): lane (l16, half) reads row l16,
    // times t0 + kc*4 + 2*half + {0,1}  -> one float2 per chunk.
    const size_t aoff = (size_t)l16 * T_LEN + (size_t)(2 * half);

    // D/C layout: VGPR v <-> rows v (half 0) / v+8 (half 1), lane%16 = time.
    const size_t doff = (size_t)(8 * half) * T_LEN + (size_t)l16;

    float carry[8];
    #pragma unroll
    for (int v = 0; v < 8; ++v) carry[v] = 0.0f;

    for (int t0 = 0; t0 < T_LEN; t0 += 16) {
        // read-stream prefetch, 8 tiles ahead (one line per row)
        if (t0 + 128 < T_LEN) {
            __builtin_prefetch(xrow + aoff + (size_t)(t0 + 128), 0, 1);
        }

        // ---- load A chunks (b64 per lane, contiguous within a row) ------
        v2f a0 = *reinterpret_cast<const v2f*>(xrow + aoff + (size_t)t0 + 0);
        v2f a1 = *reinterpret_cast<const v2f*>(xrow + aoff + (size_t)t0 + 4);
        v2f a2 = *reinterpret_cast<const v2f*>(xrow + aoff + (size_t)t0 + 8);
        v2f a3 = *reinterpret_cast<const v2f*>(xrow + aoff + (size_t)t0 + 12);

        // ---- chunked-scan matmul: D = X * L  (4 chained K=4 WMMAs) ------
        v8f c = {};
        c = __builtin_amdgcn_wmma_f32_16x16x4_f32(false, a0, false, bmat[0],
                                                  (short)0, c, false, false);
        c = __builtin_amdgcn_wmma_f32_16x16x4_f32(false, a1, false, bmat[1],
                                                  (short)0, c, false, false);
        c = __builtin_amdgcn_wmma_f32_16x16x4_f32(false, a2, false, bmat[2],
                                                  (short)0, c, false, false);
        c = __builtin_amdgcn_wmma_f32_16x16x4_f32(false, a3, false, bmat[3],
                                                  (short)0, c, false, false);

        // ---- carry add, pointwise PCEN, NT store ------------------------
        #pragma unroll
        for (int v = 0; v < 8; ++v) {
            float m = c[v] + apc * carry[v];
            // next-tile carry: M at last time of tile, rows v / v+8
            carry[v] = __shfl(m, 15, 16);

            const size_t eidx = doff + (size_t)v * T_LEN + (size_t)t0;
            float xe = xrow[eidx];

            float lg = LOG2F(EPS_F + m);                 // log2(EPS+M)
            float r  = xe * EXP2F(-alpha * lg) + bias;   // x*(EPS+M)^-a + b
            float o  = EXP2F(power * LOG2F(r)) - bias_pow;

            __builtin_nontemporal_store(o, &orow[eidx]);
        }
    }
}

extern "C" void kernel_launch(void* const* d_in, const int* in_sizes, int n_in,
                              void* d_out, int out_size, void* d_ws, size_t ws_size,
                              hipStream_t stream) {
    const float* x     = (const float*)d_in[0];
    const float* alpha = (const float*)d_in[1];
    const float* power = (const float*)d_in[2];
    const float* bias  = (const float*)d_in[3];
    float* out = (float*)d_out;

    const int nrows = in_sizes[0] / T_LEN;        // 16384 rows of length 2048
    const int waves = (nrows + 15) / 16;          // 16 rows per wave
    const int blocks = (waves + 7) / 8;           // 8 waves (256 thr) per block

    pcen_wmma_kernel<<<blocks, 256, 0, stream>>>(x, alpha, power, bias, out, nrows);
}